// TemporalGNN_51702816309731
// MI455X (gfx1250) — compile-verified
//
#include <hip/hip_runtime.h>

#define NNODES 50000
#define NEDGES 800000
#define FIN 8
#define PER 12
#define HIDN 32
#define NB 2
#define OUTC 24
#define ROWS (NB*NNODES)     /* 100000 node-batch rows */
#define NFEAT (FIN*PER)      /* 96 floats per node */

typedef float v2f __attribute__((ext_vector_type(2)));
typedef float v8f __attribute__((ext_vector_type(8)));

// ---------------- fp32 WMMA helper (16x16x4, D = A*B + C) ----------------
__device__ __forceinline__ v8f wmma4(v2f a, v2f b, v8f c) {
  return __builtin_amdgcn_wmma_f32_16x16x4_f32(false, a, false, b, (short)0, c, false, false);
}

// A-fragment from a 16x32 row-major LDS buffer (activations), K-block kb.
// Layout (ISA 7.12.2, 32-bit A 16x4): lane = M (0..15 repeated), VGPR0 = K = {0 | 2}, VGPR1 = K = {1 | 3}
__device__ __forceinline__ v2f ldsA(const float* buf, int m, int hi, int kb) {
  int k0 = kb * 4 + hi * 2;
  v2f a;
  a.x = buf[m * HIDN + k0];
  a.y = buf[m * HIDN + k0 + 1];
  return a;
}

// B-fragment from a Kx32 row-major LDS weight buffer, K-block kb, N-tile nt.
// B rows (K) striped across lane halves, N across lanes (mirror of A layout).
__device__ __forceinline__ v2f ldsB(const float* W, int m, int hi, int kb, int nt) {
  int k0 = kb * 4 + hi * 2;
  int n = nt * 16 + m;
  v2f b;
  b.x = W[k0 * HIDN + n];
  b.y = W[(k0 + 1) * HIDN + n];
  return b;
}

// Write a 16x32 C-layout pair of tiles into a row-major LDS buffer.
// C layout: VGPR r -> M = r + 8*hi, N = (lane&15) + 16*nt
__device__ __forceinline__ void storeTile(float* buf, int m, int hi, const v8f& c0, const v8f& c1) {
#pragma unroll
  for (int r = 0; r < 8; r++) {
    buf[(r + 8 * hi) * HIDN + m]      = c0[r];
    buf[(r + 8 * hi) * HIDN + 16 + m] = c1[r];
  }
}

// (16x8) @ (8x32) + bias, A fragments given in registers
__device__ __forceinline__ void gemm8(v2f ax0, v2f ax1, const float* W, const float* bias,
                                      int m, int hi, v8f& c0, v8f& c1) {
  float b0v = bias[m], b1v = bias[16 + m];
#pragma unroll
  for (int r = 0; r < 8; r++) { c0[r] = b0v; c1[r] = b1v; }
  c0 = wmma4(ax0, ldsB(W, m, hi, 0, 0), c0);
  c0 = wmma4(ax1, ldsB(W, m, hi, 1, 0), c0);
  c1 = wmma4(ax0, ldsB(W, m, hi, 0, 1), c1);
  c1 = wmma4(ax1, ldsB(W, m, hi, 1, 1), c1);
}

// concat(A1[16x32], A2[16x32]) @ L[64x32] + bias
__device__ __forceinline__ void gemm64(const float* A1, const float* A2, const float* L,
                                       const float* bias, int m, int hi, v8f& c0, v8f& c1) {
  float b0v = bias[m], b1v = bias[16 + m];
#pragma unroll
  for (int r = 0; r < 8; r++) { c0[r] = b0v; c1[r] = b1v; }
#pragma unroll
  for (int kb = 0; kb < 8; kb++) {
    v2f a = ldsA(A1, m, hi, kb);
    c0 = wmma4(a, ldsB(L, m, hi, kb, 0), c0);
    c1 = wmma4(a, ldsB(L, m, hi, kb, 1), c1);
  }
#pragma unroll
  for (int kb = 0; kb < 8; kb++) {
    v2f a = ldsA(A2, m, hi, kb);
    c0 = wmma4(a, ldsB(L, m, hi, kb + 8, 0), c0);
    c1 = wmma4(a, ldsB(L, m, hi, kb + 8, 1), c1);
  }
}

__device__ __forceinline__ float sigm(float x) { return 1.0f / (1.0f + __expf(-x)); }
__device__ __forceinline__ float tanhc(float x) {
  x = fminf(fmaxf(x, -15.f), 15.f);
  float e = __expf(2.0f * x);
  return (e - 1.0f) / (e + 1.0f);
}

// ---------------- Phase 1: GCN normalization + one-shot feature aggregation ----------------
__global__ void k_deg_init(float* deg) {
  int i = blockIdx.x * 256 + threadIdx.x;
  if (i < NNODES) deg[i] = 1.0f;                   // self-loop
}
__global__ void k_deg_acc(const int* __restrict__ ei, float* deg) {
  int e = blockIdx.x * 256 + threadIdx.x;
  if (e < NEDGES) atomicAdd(&deg[ei[NEDGES + e]], 1.0f);
}
__global__ void k_rsqrt(float* deg) {
  int i = blockIdx.x * 256 + threadIdx.x;
  if (i < NNODES) deg[i] = rsqrtf(deg[i]);         // deg >= 1 always
}
// Xagg = dinv[n]^2 * x  (self-loop term), vectorized float4
__global__ void k_selfloop(const float* __restrict__ x, const float* __restrict__ dinv,
                           float* __restrict__ xagg) {
  int t = blockIdx.x * 256 + threadIdx.x;          // over ROWS*24 float4 groups
  if (t >= ROWS * 24) return;
  int i = t / 24;
  int n = i % NNODES;
  float w = dinv[n] * dinv[n];
  float4 xv = ((const float4*)x)[t];
  float4 o; o.x = xv.x * w; o.y = xv.y * w; o.z = xv.z * w; o.w = xv.w * w;
  ((float4*)xagg)[t] = o;
}
// Xagg[b,dst,:] += norm_e * x[b,src,:]  (all 96 features, both batches; L2-resident atomics)
__global__ void k_scatter(const float* __restrict__ x, const int* __restrict__ ei,
                          const float* __restrict__ dinv, float* __restrict__ xagg) {
  int t = blockIdx.x * 256 + threadIdx.x;          // E * 24 threads, 8 floats each
  if (t >= NEDGES * 24) return;
  int e = t / 24;
  int r = t - e * 24;
  int b = r / 12;
  int c = r - b * 12;
  int s = ei[e];
  int d = ei[NEDGES + e];
  float w = dinv[s] * dinv[d];
  const float* xp = x + (size_t)(b * NNODES + s) * NFEAT + c * 8;
  float* op = xagg + (size_t)(b * NNODES + d) * NFEAT + c * 8;
#pragma unroll
  for (int j = 0; j < 8; j++) atomicAdd(op + j, w * xp[j]);
}

// ---------------- Phase 2: per-node A3TGCN GRU with fp32 WMMA ----------------
__global__ void __launch_bounds__(256) k_gru(
    const float* __restrict__ xagg, const float* __restrict__ att,
    const float* __restrict__ Wz, const float* __restrict__ bz,
    const float* __restrict__ Wr, const float* __restrict__ br,
    const float* __restrict__ Wh, const float* __restrict__ bh,
    const float* __restrict__ Lz, const float* __restrict__ lbz,
    const float* __restrict__ Lr, const float* __restrict__ lbr,
    const float* __restrict__ Lh, const float* __restrict__ lbh,
    const float* __restrict__ Wo, const float* __restrict__ bo,
    float* __restrict__ out) {
  __shared__ float sW[3 * 8 * 32];     // Wz,Wr,Wh
  __shared__ float sb[3 * 32];
  __shared__ float sL[3 * 64 * 32];    // Lz,Lr,Lh
  __shared__ float slb[3 * 32];
  __shared__ float sWo[32 * 32];       // Wout padded to 32 cols
  __shared__ float sbo[32];
  __shared__ float sG[8][16 * 32];     // per-wave staging (C-layout -> A-layout shuffle)
  __shared__ float sH[8][16 * 32];
  __shared__ float sHR[8][16 * 32];

  int tid = threadIdx.x;
  for (int t = tid; t < 8 * 32; t += 256) { sW[t] = Wz[t]; sW[256 + t] = Wr[t]; sW[512 + t] = Wh[t]; }
  for (int t = tid; t < 32; t += 256) {
    sb[t] = bz[t]; sb[32 + t] = br[t]; sb[64 + t] = bh[t];
    slb[t] = lbz[t]; slb[32 + t] = lbr[t]; slb[64 + t] = lbh[t];
    sbo[t] = (t < OUTC) ? bo[t] : 0.f;
  }
  for (int t = tid; t < 64 * 32; t += 256) { sL[t] = Lz[t]; sL[2048 + t] = Lr[t]; sL[4096 + t] = Lh[t]; }
  for (int t = tid; t < 32 * 32; t += 256) {
    int k = t >> 5, j = t & 31;
    sWo[t] = (j < OUTC) ? Wo[k * OUTC + j] : 0.f;
  }

  int wave = tid >> 5, lane = tid & 31, m = lane & 15, hi = lane >> 4;
  float* mG = sG[wave];
  float* mH = sH[wave];
  float* mHR = sHR[wave];
  for (int idx = lane; idx < 16 * 32; idx += 32) mH[idx] = 0.f;

  // temporal softmax (redundantly per thread; 12 cached loads)
  float pr[PER];
  {
    float a[PER]; float mx = -1e30f;
    for (int p = 0; p < PER; p++) { a[p] = att[p]; mx = fmaxf(mx, a[p]); }
    float sm = 0.f;
    for (int p = 0; p < PER; p++) { a[p] = __expf(a[p] - mx); sm += a[p]; }
    float inv = 1.0f / sm;
    for (int p = 0; p < PER; p++) pr[p] = a[p] * inv;
  }

  int tile = blockIdx.x * 8 + wave;                // 16 node-batch rows per wave
  long ia = (long)tile * 16 + m;
  if (ia >= ROWS) ia = ROWS - 1;                   // tail waves compute on clamped rows, skip stores
  const float* xrow = xagg + ia * (long)NFEAT;

  v8f H0, H1, A0, A1;
  for (int r = 0; r < 8; r++) { H0[r] = 0.f; H1[r] = 0.f; A0[r] = 0.f; A1[r] = 0.f; }

  __syncthreads();

  for (int p = 0; p < PER; p++) {
    // shared A fragments of Xagg_p (16x8) for all three gates
    int f0 = hi * 2;
    v2f ax0, ax1;
    ax0.x = xrow[(f0)*PER + p];     ax0.y = xrow[(f0 + 1) * PER + p];
    ax1.x = xrow[(f0 + 4) * PER + p]; ax1.y = xrow[(f0 + 5) * PER + p];

    v8f g0, g1, z0, z1, r0, r1, t0, t1;

    gemm8(ax0, ax1, sW + 0, sb + 0, m, hi, g0, g1);            // Gz = Xagg@Wz + bz
    storeTile(mG, m, hi, g0, g1);
    __syncthreads();
    gemm64(mG, mH, sL + 0, slb + 0, m, hi, z0, z1);            // [Gz|H]@Lz + lbz
    for (int r = 0; r < 8; r++) { z0[r] = sigm(z0[r]); z1[r] = sigm(z1[r]); }
    __syncthreads();

    gemm8(ax0, ax1, sW + 256, sb + 32, m, hi, g0, g1);         // Gr
    storeTile(mG, m, hi, g0, g1);
    __syncthreads();
    gemm64(mG, mH, sL + 2048, slb + 32, m, hi, r0, r1);        // [Gr|H]@Lr + lbr
    for (int r = 0; r < 8; r++) { r0[r] = sigm(r0[r]); r1[r] = sigm(r1[r]); }

    gemm8(ax0, ax1, sW + 512, sb + 64, m, hi, g0, g1);         // Gh
    __syncthreads();
    storeTile(mG, m, hi, g0, g1);
    {
      v8f q0, q1;
      for (int r = 0; r < 8; r++) { q0[r] = H0[r] * r0[r]; q1[r] = H1[r] * r1[r]; }
      storeTile(mHR, m, hi, q0, q1);                           // H * R
    }
    __syncthreads();
    gemm64(mG, mHR, sL + 4096, slb + 64, m, hi, t0, t1);       // [Gh|H*R]@Lh + lbh
    for (int r = 0; r < 8; r++) { t0[r] = tanhc(t0[r]); t1[r] = tanhc(t1[r]); }

    for (int r = 0; r < 8; r++) {                              // GRU update + attention accum
      H0[r] = z0[r] * H0[r] + (1.0f - z0[r]) * t0[r];
      H1[r] = z1[r] * H1[r] + (1.0f - z1[r]) * t1[r];
      A0[r] += pr[p] * H0[r];
      A1[r] += pr[p] * H1[r];
    }
    storeTile(mH, m, hi, H0, H1);                              // publish H for next period
    __syncthreads();
  }

  // out = relu(acc) @ Wout + bout
  for (int r = 0; r < 8; r++) { A0[r] = fmaxf(A0[r], 0.f); A1[r] = fmaxf(A1[r], 0.f); }
  storeTile(mG, m, hi, A0, A1);
  __syncthreads();
  v8f o0, o1;
  {
    float b0v = sbo[m], b1v = sbo[16 + m];
    for (int r = 0; r < 8; r++) { o0[r] = b0v; o1[r] = b1v; }
#pragma unroll
    for (int kb = 0; kb < 8; kb++) {
      v2f a = ldsA(mG, m, hi, kb);
      o0 = wmma4(a, ldsB(sWo, m, hi, kb, 0), o0);
      o1 = wmma4(a, ldsB(sWo, m, hi, kb, 1), o1);
    }
  }
#pragma unroll
  for (int r = 0; r < 8; r++) {
    long row = (long)tile * 16 + r + 8 * hi;
    if (row < ROWS) {
      out[row * OUTC + m] = o0[r];                 // cols 0..15
      if (m < 8) out[row * OUTC + 16 + m] = o1[r]; // cols 16..23
    }
  }
}

extern "C" void kernel_launch(void* const* d_in, const int* in_sizes, int n_in,
                              void* d_out, int out_size, void* d_ws, size_t ws_size,
                              hipStream_t stream) {
  (void)in_sizes; (void)n_in; (void)out_size; (void)ws_size;
  const float* x   = (const float*)d_in[0];
  const int*   ei  = (const int*)d_in[1];
  const float* att = (const float*)d_in[2];
  const float* Wz  = (const float*)d_in[3];  const float* bz  = (const float*)d_in[4];
  const float* Wr  = (const float*)d_in[5];  const float* br  = (const float*)d_in[6];
  const float* Wh  = (const float*)d_in[7];  const float* bh  = (const float*)d_in[8];
  const float* Lz  = (const float*)d_in[9];  const float* lbz = (const float*)d_in[10];
  const float* Lr  = (const float*)d_in[11]; const float* lbr = (const float*)d_in[12];
  const float* Lh  = (const float*)d_in[13]; const float* lbh = (const float*)d_in[14];
  const float* Wo  = (const float*)d_in[15]; const float* bo  = (const float*)d_in[16];
  float* out = (float*)d_out;

  float* wsf  = (float*)d_ws;
  float* deg  = wsf;                 // N floats (becomes dinv in place)
  float* xagg = wsf + 65536;         // ROWS * 96 floats = 38.4 MB

  k_deg_init<<<(NNODES + 255) / 256, 256, 0, stream>>>(deg);
  k_deg_acc<<<(NEDGES + 255) / 256, 256, 0, stream>>>(ei, deg);
  k_rsqrt<<<(NNODES + 255) / 256, 256, 0, stream>>>(deg);
  k_selfloop<<<(ROWS * 24 + 255) / 256, 256, 0, stream>>>(x, deg, xagg);
  k_scatter<<<(NEDGES * 24 + 255) / 256, 256, 0, stream>>>(x, ei, deg, xagg);

  int tiles = ROWS / 16;                       // 6250
  int blocks = (tiles + 7) / 8;                // 782
  k_gru<<<blocks, 256, 0, stream>>>(xagg, att, Wz, bz, Wr, br, Wh, bh,
                                    Lz, lbz, Lr, lbr, Lh, lbh, Wo, bo, out);
}